// GeometryEncoder_8203387535652
// MI455X (gfx1250) — compile-verified
//
#include <hip/hip_runtime.h>

typedef __attribute__((ext_vector_type(2))) float v2f;
typedef __attribute__((ext_vector_type(8))) float v8f;

#define N_QUERY        32768
#define N_BOUNDARY     4096
#define NB_TILES       (N_BOUNDARY / 16)            // 256 boundary tiles of 16
#define WAVES_PER_BLK  8
#define BLOCK_THREADS  (WAVES_PER_BLK * 32)         // 256
#define Q_PER_WAVE     32                           // two 16-row A tiles per wave
#define Q_PER_BLOCK    (WAVES_PER_BLK * Q_PER_WAVE) // 256
#define NUM_BLOCKS     (N_QUERY / Q_PER_BLOCK)      // 128
#define B128_CHUNKS    (N_BOUNDARY * 8 / 16)        // 2048 16B async copies

// min across the 16 lanes of each wave32 half via ds_swizzle (SWAPX1/2/4/8).
__device__ __forceinline__ float half_lane_min(float m) {
  m = fminf(m, __builtin_bit_cast(float,
        __builtin_amdgcn_ds_swizzle(__builtin_bit_cast(int, m), 0x041F)));
  m = fminf(m, __builtin_bit_cast(float,
        __builtin_amdgcn_ds_swizzle(__builtin_bit_cast(int, m), 0x081F)));
  m = fminf(m, __builtin_bit_cast(float,
        __builtin_amdgcn_ds_swizzle(__builtin_bit_cast(int, m), 0x101F)));
  m = fminf(m, __builtin_bit_cast(float,
        __builtin_amdgcn_ds_swizzle(__builtin_bit_cast(int, m), 0x201F)));
  return m;
}

// Reduce one 16x16 min-tile and store rows [rowbase, rowbase+16).
__device__ __forceinline__ void writeback(const float* __restrict__ x,
                                          float* __restrict__ out,
                                          int rowbase, bool up, int l, v8f mins) {
#pragma unroll
  for (int r = 0; r < 8; ++r) mins[r] = half_lane_min(mins[r]);
  // Now lanes 0-15 hold row mins M=0..7 in mins[0..7]; lanes 16-31 hold M=8..15.
  if (l < 8) {
    float v = mins[0];
#pragma unroll
    for (int r = 1; r < 8; ++r) if (l == r) v = mins[r];
    const int row   = rowbase + (up ? 8 : 0) + l;
    const float2 qr = ((const float2*)x)[row];
    out[row * 3 + 0] = qr.x;
    out[row * 3 + 1] = qr.y;
    out[row * 3 + 2] = sqrtf(fmaxf(v, 0.0f));   // clamp cancellation residue
  }
}

// dist^2(q,b) = qx^2+qy^2 - 2(qx*bx+qy*by) + bx^2+by^2
// K=4 dot: [qx, qy, qn, 1] . [-2bx, -2by, 1, bn] -> dist^2
// One V_WMMA_F32_16X16X4_F32 = a full 16x16 tile of squared distances.
__global__ __launch_bounds__(BLOCK_THREADS)
void geo_encoder_wmma(const float* __restrict__ x,
                      const float* __restrict__ bpts,
                      float* __restrict__ out) {
  __shared__ float2 rawb[N_BOUNDARY];    // 32 KB, filled by async DMA
  __shared__ v2f    bfrag[NB_TILES * 32]; // 64 KB, WMMA-B-layout fragments

  const int  tid  = threadIdx.x;
  const int  lane = tid & 31;
  const int  wave = tid >> 5;
  const int  l    = lane & 15;
  const bool up   = lane >= 16;

  // ---- Async DMA: raw boundary points global -> LDS (ASYNCcnt path) ----
  {
    const unsigned long long gbase = (unsigned long long)(uintptr_t)bpts;
    const unsigned int       lbase = (unsigned int)(uintptr_t)&rawb[0];
    for (int i = tid; i < B128_CHUNKS; i += BLOCK_THREADS) {
      unsigned int ldsaddr = lbase + (unsigned int)i * 16u;
      unsigned int goff    = (unsigned int)i * 16u;
      asm volatile("global_load_async_to_lds_b128 %0, %1, %2 offset:0"
                   :: "v"(ldsaddr), "v"(goff), "s"(gbase) : "memory");
    }
    asm volatile("s_wait_asynccnt 0x0" ::: "memory");
  }
  __syncthreads();

  // ---- Build pre-swizzled B fragments (per tile j, per lane k):
  //  k in [0,16):  (K=0,K=1) for col=k    -> (-2bx, -2by)
  //  k in [16,32): (K=2,K=3) for col=k-16 -> (1.0, bx^2+by^2)
  for (int idx = tid; idx < NB_TILES * 32; idx += BLOCK_THREADS) {
    const int j   = idx >> 5;
    const int k   = idx & 31;
    const int col = k & 15;
    const float2 b = rawb[j * 16 + col];
    v2f f;
    if (k < 16) { f.x = -2.0f * b.x; f.y = -2.0f * b.y; }
    else        { f.x = 1.0f;       f.y = b.x * b.x + b.y * b.y; }
    bfrag[idx] = f;
  }
  __syncthreads();

  // ---- Two A fragments (32 queries) per wave (ISA 32-bit A 16x4 layout) ----
  const int    qbase = blockIdx.x * Q_PER_BLOCK + wave * Q_PER_WAVE;
  const float2 q0 = ((const float2*)x)[qbase + l];
  const float2 q1 = ((const float2*)x)[qbase + 16 + l];
  v2f a0, a1;
  a0.x = up ? (q0.x * q0.x + q0.y * q0.y) : q0.x;  // K=0 lo / K=2 hi
  a0.y = up ? 1.0f                        : q0.y;  // K=1 lo / K=3 hi
  a1.x = up ? (q1.x * q1.x + q1.y * q1.y) : q1.x;
  a1.y = up ? 1.0f                        : q1.y;

  v8f mins0, mins1;
#pragma unroll
  for (int r = 0; r < 8; ++r) { mins0[r] = 3.4e38f; mins1[r] = 3.4e38f; }

  // ---- Main loop: 256 boundary tiles, 2 WMMAs per 8B LDS read ----
  for (int j = 0; j < NB_TILES; ++j) {
    const v2f b = bfrag[j * 32 + lane];
    v8f c0 = {}, c1 = {};
    c0 = __builtin_amdgcn_wmma_f32_16x16x4_f32(false, a0, false, b,
                                               (short)0, c0, false, false);
    c1 = __builtin_amdgcn_wmma_f32_16x16x4_f32(false, a1, false, b,
                                               (short)0, c1, false, false);
#pragma unroll
    for (int r = 0; r < 8; ++r) {
      mins0[r] = fminf(mins0[r], c0[r]);   // -> v_min3 with unroll fusion
      mins1[r] = fminf(mins1[r], c1[r]);
    }
  }

  // ---- Reduce + write both 16-row tiles ----
  writeback(x, out, qbase,      up, l, mins0);
  writeback(x, out, qbase + 16, up, l, mins1);
}

extern "C" void kernel_launch(void* const* d_in, const int* in_sizes, int n_in,
                              void* d_out, int out_size, void* d_ws, size_t ws_size,
                              hipStream_t stream) {
  const float* x    = (const float*)d_in[0];
  const float* bpts = (const float*)d_in[1];
  float* out        = (float*)d_out;
  geo_encoder_wmma<<<NUM_BLOCKS, BLOCK_THREADS, 0, stream>>>(x, bpts, out);
}